// OhemCELoss_67516885893515
// MI455X (gfx1250) — compile-verified
//
#include <hip/hip_runtime.h>
#include <hip/hip_bf16.h>
#include <stdint.h>

// Problem geometry (fixed by the reference)
static constexpr int   C      = 19;
static constexpr int   PLANE  = 512 * 512;          // 262144 pixels per image
static constexpr int   NIMG   = 16;
static constexpr int   P      = NIMG * PLANE;       // 4194304 pixels total
static constexpr unsigned NMIN = 262144;            // top-k count
static constexpr float THRESH_NL = 0.35667494393873245f;  // -log(0.7)
static constexpr int   TPB    = 256;
static constexpr int   NBLK   = P / TPB;            // 16384 blocks, each fully inside one image

// ---- workspace layout (bytes) -----------------------------------------------
static constexpr size_t OFF_HIST1 = 0;
static constexpr size_t OFF_HIST2 = 256 * 1024;
static constexpr size_t OFF_STATE = 512 * 1024;
static constexpr size_t OFF_PSUM  = 528384;
static constexpr size_t OFF_PCNT  = 659456;
static constexpr size_t OFF_PTOP  = 724992;
static constexpr size_t OFF_LOSS  = 1024 * 1024;

// ---- CDNA5 async global->LDS staging ---------------------------------------
typedef int v4i __attribute__((vector_size(16)));
typedef __attribute__((address_space(1))) v4i* gv4i_p;   // global
typedef __attribute__((address_space(3))) v4i* lv4i_p;   // LDS

__device__ __forceinline__ void async_copy16(const float* g, float* l) {
#if __has_builtin(__builtin_amdgcn_global_load_async_to_lds_b128)
  __builtin_amdgcn_global_load_async_to_lds_b128(
      (gv4i_p)(void*)g, (lv4i_p)(void*)l, 0, 0);
#else
  *(float4*)l = *(const float4*)g;   // synchronous fallback
#endif
}

__device__ __forceinline__ void wait_async0() {
#if __has_builtin(__builtin_amdgcn_s_wait_asynccnt)
  __builtin_amdgcn_s_wait_asynccnt(0);
#else
  asm volatile("s_wait_asynccnt 0" ::: "memory");
#endif
}

// ---- Pass 1: losses + coarse histogram + thresh partial sums ----------------
__global__ __launch_bounds__(TPB) void loss_hist_kernel(
    const float* __restrict__ logits, const int* __restrict__ labels,
    float* __restrict__ loss, unsigned* __restrict__ hist1,
    double* __restrict__ pSum, unsigned* __restrict__ pCnt)
{
  __shared__ __align__(16) float tile[C * TPB];   // 19 KB channel-major tile
  __shared__ double sred[TPB];
  __shared__ unsigned cred[TPB];

  const int t  = threadIdx.x;
  const size_t p0 = (size_t)blockIdx.x * TPB;     // first pixel of this block
  const int n  = (int)(p0 / PLANE);
  const int r0 = (int)(p0 % PLANE);
  const float* base = logits + (size_t)n * C * PLANE + r0;

  // Stage 19 x 256 floats (19 KB) into LDS with async b128 copies.
  // 1216 chunks of 16B; each of 256 threads handles up to 5 chunks.
  for (int k = t; k < C * 64; k += TPB) {
    const int c = k >> 6;        // channel
    const int j = k & 63;        // 16-byte chunk within the 256-float row
    async_copy16(base + (size_t)c * PLANE + j * 4, &tile[c * TPB + j * 4]);
  }
  wait_async0();
  __syncthreads();

  const int lab = labels[p0 + t];
  float m = tile[t];
#pragma unroll
  for (int c = 1; c < C; ++c) m = fmaxf(m, tile[c * TPB + t]);
  float s = 0.f;
#pragma unroll
  for (int c = 0; c < C; ++c) s += __expf(tile[c * TPB + t] - m);
  const float lse = m + __logf(s);
  const float l = lse - tile[lab * TPB + t];      // = -log_softmax[label] >= 0
  const bool valid = (lab != 1);

  loss[p0 + t] = valid ? l : -1.0f;
  if (valid) {
    const unsigned bits = __float_as_uint(l);
    atomicAdd(&hist1[bits >> 16], 1u);
  }
  const bool gt = valid && (l > THRESH_NL);
  sred[t] = gt ? (double)l : 0.0;
  cred[t] = gt ? 1u : 0u;
  __syncthreads();
  for (int off = TPB / 2; off > 0; off >>= 1) {
    if (t < off) { sred[t] += sred[t + off]; cred[t] += cred[t + off]; }
    __syncthreads();
  }
  if (t == 0) { pSum[blockIdx.x] = sred[0]; pCnt[blockIdx.x] = cred[0]; }
}

// ---- Radix-select scan over a 65536-bin histogram ---------------------------
// state: [0]=bin1 [1]=cntAbove1 [2]=found1 [3]=kthBits [4]=cntGT [5]=found2
__global__ __launch_bounds__(TPB) void scan_kernel(
    const unsigned* __restrict__ hist, unsigned* __restrict__ state, int pass)
{
  __shared__ unsigned csum[TPB];
  __shared__ unsigned cbins[TPB];
  __shared__ unsigned s_target;
  __shared__ unsigned s_cum;
  __shared__ int s_sel;

  const int t = threadIdx.x;
  if (pass == 2 && state[2] == 0) {
    if (t == 0) { state[5] = 0; state[3] = 0xFF800000u; state[4] = 0; }
    return;
  }
  if (t == 0) {
    s_target = (pass == 1) ? NMIN : (NMIN - state[1]);
    s_sel = -1;
  }
  __syncthreads();

  unsigned sum = 0;
  const unsigned* hp = hist + t * 256;
  for (int i = 0; i < 256; ++i) sum += hp[i];
  csum[t] = sum;
  __syncthreads();

  if (t == 0) {
    unsigned cum = 0; int sel = -1;
    for (int ct = 255; ct >= 0; --ct) {
      if (cum + csum[ct] >= s_target) { sel = ct; break; }
      cum += csum[ct];
    }
    s_sel = sel; s_cum = cum;
  }
  __syncthreads();

  const int sel = s_sel;
  if (sel < 0) {                       // fewer valid values than target
    if (t == 0) {
      if (pass == 1) state[2] = 0;
      else { state[5] = 0; state[3] = 0xFF800000u; state[4] = 0; }
    }
    return;
  }
  cbins[t] = hist[sel * 256 + t];
  __syncthreads();
  if (t == 0) {
    unsigned cum = s_cum; int b = 0;
    for (int i = 255; i >= 0; --i) {
      if (cum + cbins[i] >= s_target) { b = i; break; }
      cum += cbins[i];
    }
    const unsigned bin = (unsigned)(sel * 256 + b);
    if (pass == 1) { state[0] = bin; state[1] = cum; state[2] = 1; }
    else {
      state[3] = (state[0] << 16) | bin;   // exact float bit pattern of kth
      state[4] = state[1] + cum;           // count strictly > kth
      state[5] = 1;
    }
  }
}

// ---- Fine histogram of low 16 bits within candidate bin ---------------------
__global__ __launch_bounds__(TPB) void hist2_kernel(
    const float* __restrict__ loss, const unsigned* __restrict__ state,
    unsigned* __restrict__ hist2)
{
  if (state[2] == 0) return;
  const unsigned bin1 = state[0];
  const int i = blockIdx.x * TPB + threadIdx.x;
  const float l = loss[i];
  if (l < 0.f) return;
  const unsigned bits = __float_as_uint(l);
  if ((bits >> 16) == bin1) atomicAdd(&hist2[bits & 0xFFFFu], 1u);
}

// ---- Sum of losses strictly greater than kth --------------------------------
__global__ __launch_bounds__(TPB) void topsum_kernel(
    const float* __restrict__ loss, const unsigned* __restrict__ state,
    double* __restrict__ pTop)
{
  __shared__ double sred[TPB];
  const int t = threadIdx.x;
  const int i = blockIdx.x * TPB + t;
  const float T = __uint_as_float(state[3]);
  const float l = loss[i];
  sred[t] = (l >= 0.f && l > T) ? (double)l : 0.0;
  __syncthreads();
  for (int off = TPB / 2; off > 0; off >>= 1) {
    if (t < off) sred[t] += sred[t + off];
    __syncthreads();
  }
  if (t == 0) pTop[blockIdx.x] = sred[0];
}

// ---- Final deterministic reduction + branch select --------------------------
__global__ __launch_bounds__(TPB) void finalize_kernel(
    const double* __restrict__ pSum, const unsigned* __restrict__ pCnt,
    const double* __restrict__ pTop, const unsigned* __restrict__ state,
    float* __restrict__ out)
{
  __shared__ double a[TPB];
  __shared__ double b[TPB];
  __shared__ unsigned long long cc[TPB];
  const int t = threadIdx.x;
  double s1 = 0.0, s2 = 0.0; unsigned long long c0 = 0;
  for (int i = t; i < NBLK; i += TPB) { s1 += pSum[i]; s2 += pTop[i]; c0 += pCnt[i]; }
  a[t] = s1; b[t] = s2; cc[t] = c0;
  __syncthreads();
  for (int off = TPB / 2; off > 0; off >>= 1) {
    if (t < off) { a[t] += a[t + off]; b[t] += b[t + off]; cc[t] += cc[t + off]; }
    __syncthreads();
  }
  if (t == 0) {
    if (state[5] == 0) { *out = -INFINITY; return; }
    const float T = __uint_as_float(state[3]);              // exact kth value
    const unsigned cntGTk = state[4];                        // count strictly > T
    const double mean_top = (b[0] + (double)(NMIN - cntGTk) * (double)T) / (double)NMIN;
    const unsigned long long cg = cc[0] ? cc[0] : 1ull;
    const double mean_gt = a[0] / (double)cg;
    *out = (T > THRESH_NL) ? (float)mean_gt : (float)mean_top;
  }
}

extern "C" void kernel_launch(void* const* d_in, const int* in_sizes, int n_in,
                              void* d_out, int out_size, void* d_ws, size_t ws_size,
                              hipStream_t stream) {
  const float* logits = (const float*)d_in[0];
  const int*   labels = (const int*)d_in[1];
  float*       out    = (float*)d_out;
  char*        ws     = (char*)d_ws;

  unsigned* hist1 = (unsigned*)(ws + OFF_HIST1);
  unsigned* hist2 = (unsigned*)(ws + OFF_HIST2);
  unsigned* state = (unsigned*)(ws + OFF_STATE);
  double*   pSum  = (double*)  (ws + OFF_PSUM);
  unsigned* pCnt  = (unsigned*)(ws + OFF_PCNT);
  double*   pTop  = (double*)  (ws + OFF_PTOP);
  float*    loss  = (float*)   (ws + OFF_LOSS);

  // zero histograms/state/partials every call (capture-safe)
  (void)hipMemsetAsync(ws, 0, OFF_LOSS, stream);

  loss_hist_kernel<<<NBLK, TPB, 0, stream>>>(logits, labels, loss, hist1, pSum, pCnt);
  scan_kernel<<<1, TPB, 0, stream>>>(hist1, state, 1);
  hist2_kernel<<<NBLK, TPB, 0, stream>>>(loss, state, hist2);
  scan_kernel<<<1, TPB, 0, stream>>>(hist2, state, 2);
  topsum_kernel<<<NBLK, TPB, 0, stream>>>(loss, state, pTop);
  finalize_kernel<<<1, TPB, 0, stream>>>(pSum, pCnt, pTop, state, out);
}